// SparsePoolingLayer_40681930228271
// MI455X (gfx1250) — compile-verified
//
#include <hip/hip_runtime.h>

typedef float v2f __attribute__((ext_vector_type(2)));
typedef float v4f __attribute__((ext_vector_type(4)));
typedef float v8f __attribute__((ext_vector_type(8)));

#define B_      64
#define CIN_    64
#define H_      64
#define COUT_   128
#define OHW_    62
#define NPIX_   3844          // 62*62
#define NTOT_   31490048ull   // 64*128*3844
#define NTILES_ 61            // ceil(3844/64)
#define NBLK_   3904          // 61*64
#define TAU_    0.2f
#define EPS2_   1.0e-8f       // (1e-4)^2
#define MAXIT_  96
#define WPKN_   73728         // 9*128*64 repacked W_ff

// D = A(16x4,f32) * B(4x16,f32) + C, full fp32 matrix op (CDNA5 V_WMMA_F32_16X16X4_F32)
__device__ __forceinline__ v8f wmma_f32_16x16x4(v2f a, v2f b, v8f c) {
  return __builtin_amdgcn_wmma_f32_16x16x4_f32(false, a, false, b, (short)0, c,
                                               false, false);
}

// LDS tiles use a K-pair interleaved layout: elem(c, pix) at ((c>>1)*64+pix)*2+(c&1)
// so the B fragment {row kb, row kb+1} (kb even) is one contiguous b64.

// ---------------------------------------------------------------------------
// One-time: repack W_ff (o,c,kh,kw) -> (tap,o,c) so conv A-fragments are
// contiguous 64-bit loads.
// ---------------------------------------------------------------------------
__global__ __launch_bounds__(256) void lca_pack(const float* __restrict__ wff,
                                                float* __restrict__ wpk) {
  int idx = blockIdx.x * 256 + threadIdx.x;
  if (idx >= WPKN_) return;
  int tap = idx / (COUT_ * CIN_);
  int rem = idx - tap * (COUT_ * CIN_);
  int o = rem / CIN_, c = rem - o * CIN_;
  int kh = tap / 3, kw = tap - 3 * kh;
  wpk[idx] = wff[((o * CIN_ + c) * 3 + kh) * 3 + kw];
}

// ---------------------------------------------------------------------------
// Phase 1: u0 = conv3x3_valid(x, W_ff); u = 0; a = relu(u0)
// 9 tap-shifted GEMMs:  u0_tile(128o x 64pix) += Wtap(128x64) @ xs(64x64)
// One block: one image b, one 64-pixel tile, all 128 output channels.
// ---------------------------------------------------------------------------
__global__ __launch_bounds__(256) void lca_conv(
    const float* __restrict__ x, const float* __restrict__ wpk,
    float* __restrict__ u, float* __restrict__ u0, float* __restrict__ aout) {
  __shared__ float xs[CIN_ * 64];  // interleaved [c/2][pix][c%2], 16 KB

  const int tile = blockIdx.x, b = blockIdx.y;
  const int tid  = threadIdx.x;
  const int lane = tid & 31, wave = tid >> 5;     // wave32: 8 waves
  const int half = lane >> 4, lq = lane & 15;
  const int pbase = tile * 64;
  const int orow = wave * 16 + lq;                // A-fragment row (output ch)

  const v8f zero = {0.f, 0.f, 0.f, 0.f, 0.f, 0.f, 0.f, 0.f};
  v8f acc[4];
#pragma unroll
  for (int nt = 0; nt < 4; ++nt) acc[nt] = zero;

  const int pix_l = tid & 63, cgrp = tid >> 6;    // staging decomposition

  for (int tap = 0; tap < 9; ++tap) {
    const int kh = tap / 3, kw = tap - 3 * kh;
    __syncthreads();
    {  // stage shifted input tile (coalesced along pixels)
      int p  = pbase + pix_l;
      int pc = p < NPIX_ ? p : (NPIX_ - 1);       // clamp: stores are masked later
      int oh = pc / OHW_, ow = pc - OHW_ * oh;
      const float* src =
          x + ((size_t)(b * CIN_) * H_ + (oh + kh)) * H_ + (ow + kw);
#pragma unroll
      for (int cc = 0; cc < 16; ++cc) {
        int c = cgrp * 16 + cc;
        xs[((c >> 1) * 64 + pix_l) * 2 + (c & 1)] = src[(size_t)c * (H_ * H_)];
      }
    }
    __syncthreads();
    const float* wtap = wpk + ((size_t)tap * COUT_ + orow) * CIN_;
    // K = 64 input channels, 4 per WMMA
#pragma unroll 4
    for (int k0 = 0; k0 < CIN_; k0 += 4) {
      const int kb = k0 + 2 * half;               // always even
      v2f av = *(const v2f*)(wtap + kb);          // contiguous b64 A-fragment
#pragma unroll
      for (int nt = 0; nt < 4; ++nt) {
        // {xs[kb][n], xs[kb+1][n]} contiguous -> single ds_load_b64
        v2f bv = *(const v2f*)(&xs[((kb >> 1) * 64 + nt * 16 + lq) * 2]);
        acc[nt] = wmma_f32_16x16x4(av, bv, acc[nt]);
      }
    }
  }

  // epilogue: write u0, u=0, a0=relu(u0)
#pragma unroll
  for (int nt = 0; nt < 4; ++nt) {
    int p = pbase + nt * 16 + lq;
    if (p < NPIX_) {
#pragma unroll
      for (int i = 0; i < 8; ++i) {
        int o = wave * 16 + i + 8 * half;         // D layout: row i + 8*half
        size_t idx = (size_t)(b * COUT_ + o) * NPIX_ + p;
        float v = acc[nt][i];
        u0[idx]   = v;
        u[idx]    = 0.0f;
        aout[idx] = v > 0.0f ? v : 0.0f;
      }
    }
  }
}

// ---------------------------------------------------------------------------
// Phase 2 body: lateral = Wr @ a (per-pixel); u = 0.8u + 0.2(u0-lateral) in place;
// a = relu(u - thr) in place; per-block partial sums of ||du||^2, ||u||^2.
// Safe in-place: this block owns ALL channels of its pixel tile.
// ---------------------------------------------------------------------------
__global__ __launch_bounds__(256) void lca_update(
    const float* __restrict__ wr, const float* __restrict__ u0,
    const float* __restrict__ thr, float* __restrict__ u,
    float* __restrict__ a, float* __restrict__ pd2, float* __restrict__ pn2,
    const int* __restrict__ stop) {
  if (*stop) return;  // converged on an earlier pre-launched iteration

  __shared__ float as_[COUT_ * 64];  // interleaved [c/2][pix][c%2], 32 KB
  __shared__ float s_thr[COUT_];
  __shared__ float rd[256], rn[256];

  const int tile = blockIdx.x, b = blockIdx.y;
  const int tid  = threadIdx.x;
  const int lane = tid & 31, wave = tid >> 5;
  const int half = lane >> 4, lq = lane & 15;
  const int pbase = tile * 64;
  const int orow = wave * 16 + lq;

  {  // stage activation tile: channel-pair quads, b128 in / zip / b128 to LDS
    const int pgrp = (tid & 15) * 4;    // pixel quad within tile
    const int pr0  = (tid >> 4) * 4;    // 4 channel-pairs per thread
    const int p = pbase + pgrp;         // NPIX_ % 4 == 0: quads never straddle
    const v4f zv = {0.f, 0.f, 0.f, 0.f};
#pragma unroll
    for (int pp = 0; pp < 4; ++pp) {
      int pair = pr0 + pp;
      int c0 = pair * 2;
      v4f q0 = zv, q1 = zv;
      if (p < NPIX_) {
        q0 = *(const v4f*)(a + (size_t)(b * COUT_ + c0 + 0) * NPIX_ + p);
        q1 = *(const v4f*)(a + (size_t)(b * COUT_ + c0 + 1) * NPIX_ + p);
      }
      v4f r0 = {q0.x, q1.x, q0.y, q1.y};
      v4f r1 = {q0.z, q1.z, q0.w, q1.w};
      float* dst = &as_[(pair * 64 + pgrp) * 2];
      *(v4f*)(dst + 0) = r0;
      *(v4f*)(dst + 4) = r1;
    }
    if (tid < COUT_) s_thr[tid] = thr[tid];
  }
  __syncthreads();

  const v8f zero = {0.f, 0.f, 0.f, 0.f, 0.f, 0.f, 0.f, 0.f};
  v8f acc[4];
#pragma unroll
  for (int nt = 0; nt < 4; ++nt) acc[nt] = zero;

#pragma unroll 4
  for (int k0 = 0; k0 < COUT_; k0 += 4) {
    const int kb = k0 + 2 * half;                 // always even
    // Wr rows are per-wave disjoint -> straight from global (L2/WGP$-hot), b64
    v2f av = *(const v2f*)(wr + orow * COUT_ + kb);
#pragma unroll
    for (int nt = 0; nt < 4; ++nt) {
      v2f bv = *(const v2f*)(&as_[((kb >> 1) * 64 + nt * 16 + lq) * 2]);
      acc[nt] = wmma_f32_16x16x4(av, bv, acc[nt]);
    }
  }

  // fused state update + norm accumulation
  float d2 = 0.0f, n2 = 0.0f;
#pragma unroll
  for (int nt = 0; nt < 4; ++nt) {
    int p = pbase + nt * 16 + lq;
    if (p < NPIX_) {
#pragma unroll
      for (int i = 0; i < 8; ++i) {
        int o = wave * 16 + i + 8 * half;
        size_t idx = (size_t)(b * COUT_ + o) * NPIX_ + p;
        float lat = acc[nt][i];
        float uu  = u[idx];
        float un  = (1.0f - TAU_) * uu + TAU_ * (u0[idx] - lat);
        float dd  = un - uu;
        d2 += dd * dd;
        n2 += un * un;
        u[idx] = un;
        float av_ = un - s_thr[o];
        a[idx] = av_ > 0.0f ? av_ : 0.0f;
      }
    }
  }

  // deterministic block reduction -> per-block partial (no float atomics)
  rd[tid] = d2;
  rn[tid] = n2;
  __syncthreads();
  for (int s = 128; s > 0; s >>= 1) {
    if (tid < s) { rd[tid] += rd[tid + s]; rn[tid] += rn[tid + s]; }
    __syncthreads();
  }
  if (tid == 0) {
    int bid = b * NTILES_ + tile;
    pd2[bid] = rd[0];
    pn2[bid] = rn[0];
  }
}

// convergence test: stop when ||du||^2 < eps^2 * ||u||^2
__global__ __launch_bounds__(256) void lca_check(const float* __restrict__ pd2,
                                                 const float* __restrict__ pn2,
                                                 int* __restrict__ stop) {
  if (*stop) return;
  __shared__ float sd[256], sn[256];
  float d = 0.0f, n = 0.0f;
  for (int i = threadIdx.x; i < NBLK_; i += 256) { d += pd2[i]; n += pn2[i]; }
  sd[threadIdx.x] = d;
  sn[threadIdx.x] = n;
  __syncthreads();
  for (int s = 128; s > 0; s >>= 1) {
    if (threadIdx.x < s) {
      sd[threadIdx.x] += sd[threadIdx.x + s];
      sn[threadIdx.x] += sn[threadIdx.x + s];
    }
    __syncthreads();
  }
  if (threadIdx.x == 0 && sd[0] < EPS2_ * sn[0]) *stop = 1;
}

__global__ void lca_init(int* __restrict__ stop) { *stop = 0; }

// ---------------------------------------------------------------------------
extern "C" void kernel_launch(void* const* d_in, const int* in_sizes, int n_in,
                              void* d_out, int out_size, void* d_ws,
                              size_t ws_size, hipStream_t stream) {
  const float* x   = (const float*)d_in[0];  // (64,64,64,64)
  const float* wff = (const float*)d_in[1];  // (128,64,3,3)
  const float* wr  = (const float*)d_in[2];  // (128,128,1,1)
  const float* thr = (const float*)d_in[3];  // (128,)
  float* aout = (float*)d_out;               // (64,128,62,62) -> live 'a' buffer

  float* wsf = (float*)d_ws;
  float* u   = wsf;                          // membrane potential
  float* u0  = wsf + (size_t)NTOT_;          // feedforward drive
  float* pd2 = wsf + 2 * (size_t)NTOT_;      // per-block ||du||^2 partials
  float* pn2 = pd2 + 4096;                   // per-block ||u||^2  partials
  int*  stop = (int*)(pn2 + 4096);           // convergence flag (32 floats pad)
  float* wpk = pn2 + 4096 + 32;              // repacked W_ff [tap][o][c]

  dim3 grid(NTILES_, B_);
  lca_init<<<1, 1, 0, stream>>>(stop);
  lca_pack<<<(WPKN_ + 255) / 256, 256, 0, stream>>>(wff, wpk);
  lca_conv<<<grid, 256, 0, stream>>>(x, wpk, u, u0, aout);
  for (int it = 0; it < MAXIT_; ++it) {
    lca_update<<<grid, 256, 0, stream>>>(wr, u0, thr, u, aout, pd2, pn2, stop);
    lca_check<<<1, 256, 0, stream>>>(pd2, pn2, stop);
  }
}